// StepwiseVectorizedConstantVelocityModel_64364379898613
// MI455X (gfx1250) — compile-verified
//
#include <hip/hip_runtime.h>

// ---------------- constants matching the reference ----------------
#define NPTS   128
#define NSTEPS 64
#define KDIM   128                  // 2 * NSTEPS (v0 row length, flattened (c,s))
#define DELTA_F (100.0f / 63.0f)    // MAX_TIME / (STEPS-1), correctly rounded f32
#define EPS_F   1e-9f
#define HALF_SQRT_PI 0.88622692545275801365f  // sqrt(pi)/2

// ---------------- workspace layout (floats) ----------------
#define WS_U1   0                       // 64
#define WS_U2   64                      // 64
#define WS_G    128                     // 128*128 Gram matrix
#define WS_W1   (128 + 128 * 128)       // 256 : w1[n][c]
#define WS_EVP  (WS_W1 + 256)           // up to 64 event partials
#define WS_NEP  (WS_EVP + 64)           // 64 non-event partials

typedef __attribute__((ext_vector_type(2))) float v2f;
typedef __attribute__((ext_vector_type(8))) float v8f;

// ---------------- deterministic block reduction (blockDim == 256) ----------------
__device__ __forceinline__ float blockReduce256(float v, float* red) {
#pragma unroll
  for (int off = 16; off > 0; off >>= 1) v += __shfl_xor(v, off, 32);
  int w = threadIdx.x >> 5;
  if ((threadIdx.x & 31) == 0) red[w] = v;
  __syncthreads();
  float tot = 0.0f;
  if (threadIdx.x == 0) {
#pragma unroll
    for (int k = 0; k < 8; ++k) tot += red[k];
  }
  return tot;  // valid on thread 0 only
}

// ---------------- K1: per-step event moments + w1 matvec ----------------
// U1[s] = sum_{e: idx(e)=s} u_e ; U2[s] = sum u_e^2 ; u_e = dt_s + frac_e
// w1[n][c] = sum_s U1[s] * v0[n][c][s]
__global__ void k_moments(const float* __restrict__ evt, int E,
                          const float* __restrict__ v0,
                          float* __restrict__ ws) {
  __shared__ float sU1[NSTEPS];
  int t = threadIdx.x;  // 256 threads
  if (t < NSTEPS) {
    float u1 = 0.0f, u2 = 0.0f;
    float dt = (t == 0) ? 0.0f : DELTA_F;
    for (int e = 0; e < E; ++e) {
      float tv = evt[e] / DELTA_F;        // division: matches reference rounding
      float tf = floorf(tv);
      int idx = (int)tf;
      idx = idx < 0 ? 0 : (idx > NSTEPS - 1 ? NSTEPS - 1 : idx);
      if (idx == t) {
        float u = dt + (tv - tf);         // frac uses unclipped floor
        u1 += u;
        u2 += u * u;
      }
    }
    sU1[t] = u1;
    ws[WS_U1 + t] = u1;
    ws[WS_U2 + t] = u2;
  }
  __syncthreads();
  // 256 threads -> (n, c) = (t>>1, t&1)
  int n = t >> 1, c = t & 1;
  const float* row = v0 + n * KDIM + c * NSTEPS;
  float acc = 0.0f;
#pragma unroll 8
  for (int s = 0; s < NSTEPS; ++s) acc += sU1[s] * row[s];
  ws[WS_W1 + t] = acc;
}

// ---------------- K2: Gram matrix via WMMA  G = (v0 .* U2) x v0^T ----------------
// v0 viewed as M[128][128] (row n, col k=c*64+s). G[i][j] = sum_k U2[k&63]*M[i][k]*M[j][k].
// One wave (32 lanes) per 16x16 tile; K accumulated 4 at a time with V_WMMA_F32_16X16X4_F32.
__global__ void k_gram(const float* __restrict__ v0, float* __restrict__ ws) {
  const float* U2 = ws + WS_U2;
  float* G = ws + WS_G;
  int ti = blockIdx.x >> 3;
  int tj = blockIdx.x & 7;
  int lane = threadIdx.x;          // 0..31, EXEC all ones
  int m = lane & 15;               // row within tile for A; col within tile for B
  int koff = (lane >> 4) << 1;     // lanes 16-31 hold K=k0+2,k0+3

  const float* rowA = v0 + (ti * 16 + m) * KDIM;  // A[m][k] = U2[k]*M[ti*16+m][k]
  const float* rowB = v0 + (tj * 16 + m) * KDIM;  // B[k][n] = M[tj*16+n][k]

  v8f acc = {};
#pragma unroll
  for (int k0 = 0; k0 < KDIM; k0 += 4) {
    int k = k0 + koff;
    v2f a, b;
    a.x = rowA[k]     * U2[k & (NSTEPS - 1)];
    a.y = rowA[k + 1] * U2[(k + 1) & (NSTEPS - 1)];
    b.x = rowB[k];
    b.y = rowB[k + 1];
    acc = __builtin_amdgcn_wmma_f32_16x16x4_f32(
        /*neg_a=*/false, a, /*neg_b=*/false, b,
        /*c_mod=*/(short)0, acc, /*reuse_a=*/false, /*reuse_b=*/false);
  }
  // D layout: VGPR r holds M = r (lanes 0-15) / M = 8+r (lanes 16-31), N = lane&15
  int row0 = ti * 16 + ((lane >> 4) << 3);
  int col = tj * 16 + m;
#pragma unroll
  for (int r = 0; r < 8; ++r) G[(row0 + r) * KDIM + col] = acc[r];
}

// ---------------- K3: per-pair event reduction (gathers G, w1) ----------------
__global__ void k_pairs(const float* __restrict__ z0,
                        const int* __restrict__ pi, const int* __restrict__ pj,
                        int P, float Ef, const float* __restrict__ ws,
                        float* __restrict__ partials) {
  __shared__ float red[8];
  const float* G = ws + WS_G;
  const float* w1 = ws + WS_W1;
  int p = blockIdx.x * blockDim.x + threadIdx.x;
  float term = 0.0f;
  if (p < P) {
    int i = pi[p], j = pj[p];
    float dzx = z0[2 * i]     - z0[2 * j];
    float dzy = z0[2 * i + 1] - z0[2 * j + 1];
    float wx  = w1[2 * i]     - w1[2 * j];
    float wy  = w1[2 * i + 1] - w1[2 * j + 1];
    float gii = G[i * KDIM + i];
    float gjj = G[j * KDIM + j];
    float gij = G[i * KDIM + j];
    term = Ef * (dzx * dzx + dzy * dzy)
         + 2.0f * (dzx * wx + dzy * wy)
         + (gii + gjj - 2.0f * gij);
  }
  float tot = blockReduce256(term, red);
  if (threadIdx.x == 0) partials[blockIdx.x] = tot;
}

// ---------------- K4: non-event analytic integral over (i<j, s) ----------------
// 8x8 grid of 16x16 (i,j) tiles; v0 rows staged in LDS; s=0 term is exactly zero.
__global__ void k_nonevent(const float* __restrict__ z0,
                           const float* __restrict__ v0,
                           const float* __restrict__ beta,
                           float* __restrict__ partials) {
  __shared__ float Vi[16 * KDIM];
  __shared__ float Vj[16 * KDIM];
  __shared__ float Zi[32];
  __shared__ float Zj[32];
  __shared__ float red[8];

  int bi = blockIdx.x >> 3;
  int bj = blockIdx.x & 7;
  int t = threadIdx.x;
  float sum = 0.0f;

  if (bi <= bj) {  // block-uniform branch
    for (int x = t; x < 16 * KDIM; x += 256) {
      Vi[x] = v0[bi * 16 * KDIM + x];
      Vj[x] = v0[bj * 16 * KDIM + x];
    }
    if (t < 32) {
      Zi[t] = z0[bi * 32 + t];
      Zj[t] = z0[bj * 32 + t];
    }
    __syncthreads();

    int ti = t >> 4, tj = t & 15;
    int i = bi * 16 + ti, j = bj * 16 + tj;
    if (i < j) {
      float b = beta[0];
      float dz0x = Zi[2 * ti]     - Zj[2 * tj];
      float dz0y = Zi[2 * ti + 1] - Zj[2 * tj + 1];
      const float* vi = &Vi[ti * KDIM];
      const float* vj = &Vj[tj * KDIM];
#pragma unroll 4
      for (int s = 1; s < NSTEPS; ++s) {
        float dvx = vi[s]          - vj[s];
        float dvy = vi[NSTEPS + s] - vj[NSTEPS + s];
        float dzx = dz0x + DELTA_F * dvx;
        float dzy = dz0y + DELTA_F * dvy;
        float r2 = dvx * dvx + dvy * dvy + EPS_F;
        float r = sqrtf(r2);
        float zdv = dzx * dvx + dzy * dvy;
        float bshift = zdv / r2;
        float cc = dzx * dzx + dzy * dzy - zdv * zdv / r2;
        float val = expf(b - cc) * (HALF_SQRT_PI / r)
                  * (erff(r * (DELTA_F + bshift)) - erff(r * bshift));
        sum += val;
      }
    }
    __syncthreads();
  }
  float tot = blockReduce256(sum, red);
  if (threadIdx.x == 0) partials[blockIdx.x] = tot;
}

// ---------------- K5: deterministic finalize ----------------
__global__ void k_final(const float* __restrict__ beta,
                        const float* __restrict__ ws,
                        float* __restrict__ out,
                        int nEvParts, float PEf) {
  float ev = 0.0f;
  for (int k = 0; k < nEvParts; ++k) ev += ws[WS_EVP + k];
  float ne = 0.0f;
  for (int k = 0; k < 64; ++k) ne += ws[WS_NEP + k];
  // result = (beta*P*E - sum_pair_event) - non_event
  out[0] = beta[0] * PEf - ev - ne;
}

extern "C" void kernel_launch(void* const* d_in, const int* in_sizes, int n_in,
                              void* d_out, int out_size, void* d_ws, size_t ws_size,
                              hipStream_t stream) {
  const float* beta = (const float*)d_in[0];
  const float* z0   = (const float*)d_in[1];
  const float* v0   = (const float*)d_in[2];
  const float* evt  = (const float*)d_in[3];
  // d_in[4], d_in[5] = t0, tn (unused by the math beyond constants)
  const int* pi = (const int*)d_in[6];
  const int* pj = (const int*)d_in[7];

  float* out = (float*)d_out;
  float* ws  = (float*)d_ws;

  int E = in_sizes[3];
  int P = in_sizes[6];
  int evBlocks = (P + 255) / 256;
  if (evBlocks > 64) evBlocks = 64;  // workspace partial capacity guard

  hipLaunchKernelGGL(k_moments,  dim3(1),        dim3(256), 0, stream, evt, E, v0, ws);
  hipLaunchKernelGGL(k_gram,     dim3(64),       dim3(32),  0, stream, v0, ws);
  hipLaunchKernelGGL(k_pairs,    dim3(evBlocks), dim3(256), 0, stream,
                     z0, pi, pj, P, (float)E, ws, ws + WS_EVP);
  hipLaunchKernelGGL(k_nonevent, dim3(64),       dim3(256), 0, stream,
                     z0, v0, beta, ws + WS_NEP);
  hipLaunchKernelGGL(k_final,    dim3(1),        dim3(1),   0, stream,
                     beta, ws, out, evBlocks, (float)P * (float)E);
}